// MaskPruningGlobalAttentionSpatial_74887049773157
// MI455X (gfx1250) — compile-verified
//
#include <hip/hip_runtime.h>
#include <math.h>

// ---------------------------------------------------------------------------
// MaskPruningGlobalAttention, fused flash-attention style for MI455X (gfx1250)
//
// Shapes: B=2, C=64, H=W=80 -> n=6400.
//   Q = Wq*fg + bq ; K = Wk*bg + bk ; V = Wv*fg + bv     (tiny GEMMs)
//   S = Q^T K  (n x n, K-dim = 64)                        (big GEMM #1)
//   P = exp(S * mask_j)  (max-free softmax: |S|<~40, fp32 safe)
//   agg = (P / rowsum(P)) V^T                             (big GEMM #2)
//   out = fg*mask + gamma*(1-mask)*agg
//
// Never materializes the n x n matrix (~656 MB HBM traffic avoided); all
// matrix math on V_WMMA_F32_16X16X4_F32 (full fp32, softmax-safe).
// K/V tiles are double-buffered into LDS with GLOBAL_LOAD_ASYNC_TO_LDS_B128
// (ASYNCcnt), overlapping staging with the 128 WMMAs of the previous tile.
// ---------------------------------------------------------------------------

typedef __attribute__((ext_vector_type(2))) float v2f;
typedef __attribute__((ext_vector_type(8))) float v8f;

namespace {
constexpr int kC    = 64;      // channels
constexpr int kN    = 6400;    // pixels (80*80)
constexpr int kTile = 64;      // pixel tile (query rows / key cols per step)
// LDS row strides (floats), chosen for conflict-free WMMA fragment reads and
// 16B-aligned rows for async B128 copies:
//  ks: stride 72 -> cross-half bank offset 2*72 mod 64 = 16 (no lane collision)
//  vs/ss: stride 68 -> bank = 4*m16 + j; 4*dm16 never equals cross-half +2
constexpr int kSK = 72;
constexpr int kSV = 68;
constexpr int kSS = 68;
}

__device__ __forceinline__ v8f wmma_f32(v2f a, v2f b, v8f c) {
  // D = A(16x4 f32) * B(4x16 f32) + C(16x16 f32)
  return __builtin_amdgcn_wmma_f32_16x16x4_f32(
      /*neg_a=*/false, a, /*neg_b=*/false, b,
      /*c_mod=*/(short)0, c, /*reuse_a=*/false, /*reuse_b=*/false);
}

// Async copy of one 16-byte chunk global -> LDS (per lane). Tracked by
// ASYNCcnt; completion enforced with s_wait_asynccnt.
__device__ __forceinline__ void async_copy_b128(const float* gaddr,
                                                void* lds_ptr) {
  // Low 32 bits of a generic pointer to LDS are the 0-based LDS byte offset.
  unsigned lds_off = (unsigned)(uintptr_t)lds_ptr;
  asm volatile("global_load_async_to_lds_b128 %0, %1, off"
               :: "v"(lds_off), "v"(gaddr)
               : "memory");
}

__device__ __forceinline__ void wait_async_zero() {
  asm volatile("s_wait_asynccnt 0" ::: "memory");
}

// ---------------------------------------------------------------------------
// Kernel 1: QKV projection.  O[64 x 64pix] = W[64x64] * X[64 x 64pix] + bias
// grid = (100 pixel-tiles, 2 batches, 3 projections), 128 threads (4 waves).
// ---------------------------------------------------------------------------
__global__ __launch_bounds__(128)
void qkv_proj_kernel(const float* __restrict__ fg, const float* __restrict__ bg,
                     const float* __restrict__ wq, const float* __restrict__ bq,
                     const float* __restrict__ wk, const float* __restrict__ bk,
                     const float* __restrict__ wv, const float* __restrict__ bv,
                     float* __restrict__ Q, float* __restrict__ K,
                     float* __restrict__ V) {
  const int tile = blockIdx.x;
  const int b    = blockIdx.y;
  const int sel  = blockIdx.z;

  const float* X; const float* W; const float* bias; float* O;
  if (sel == 0)      { X = fg; W = wq; bias = bq; O = Q; }
  else if (sel == 1) { X = bg; W = wk; bias = bk; O = K; }
  else               { X = fg; W = wv; bias = bv; O = V; }
  X += (size_t)b * kC * kN;
  O += (size_t)b * kC * kN;

  const int p0    = tile * kTile;
  const int lane  = threadIdx.x & 31;
  const int wave  = threadIdx.x >> 5;     // 0..3, owns 16 output channels
  const int m16   = lane & 15;
  const int khalf = lane >> 4;            // K-pair selector (ISA A/B layout)
  const int mrow  = wave * 16 + m16;      // output channel for A fragment

  // A fragments: rows of W for this wave's 16 output channels, all 16 K-steps.
  v2f a[16];
#pragma unroll
  for (int kk = 0; kk < 16; ++kk) {
    const int kidx = kk * 4 + khalf * 2;
    a[kk].x = W[mrow * kC + kidx];
    a[kk].y = W[mrow * kC + kidx + 1];
  }

#pragma unroll
  for (int jt = 0; jt < 4; ++jt) {
    const int n = p0 + jt * 16 + m16;     // pixel column for B/D fragments
    v8f acc = {};
#pragma unroll
    for (int kk = 0; kk < 16; ++kk) {
      const int kidx = kk * 4 + khalf * 2;
      v2f bf;
      bf.x = X[(size_t)kidx * kN + n];
      bf.y = X[(size_t)(kidx + 1) * kN + n];
      acc = wmma_f32(a[kk], bf, acc);
    }
#pragma unroll
    for (int r = 0; r < 8; ++r) {
      const int cout = wave * 16 + r + khalf * 8;   // D layout row
      O[(size_t)cout * kN + n] = acc[r] + bias[cout];
    }
  }
}

// ---------------------------------------------------------------------------
// Kernel 2: fused attention.  grid = (100 query tiles, 2 batches),
// 128 threads (4 waves); wave w owns query rows [16w, 16w+16).
// K/V tiles double-buffered via async global->LDS copies.
// ---------------------------------------------------------------------------
__global__ __launch_bounds__(128)
void flash_attn_kernel(const float* __restrict__ Q, const float* __restrict__ K,
                       const float* __restrict__ V,
                       const float* __restrict__ mask,
                       const float* __restrict__ fg,
                       const float* __restrict__ gamma,
                       float* __restrict__ out) {
  __shared__ float ks[2][kTile][kSK];   // K tile, [c][j] (global layout)
  __shared__ float vs[2][kTile][kSV];   // V tile, [c][j] (global layout)
  __shared__ float ss[kTile][kSS];      // P tile [i][j]; reused for transpose

  const int it = blockIdx.x;
  const int b  = blockIdx.y;
  const int i0 = it * kTile;

  const float* Qb = Q + (size_t)b * kC * kN;
  const float* Kb = K + (size_t)b * kC * kN;
  const float* Vb = V + (size_t)b * kC * kN;
  const float* mb = mask + (size_t)b * kN;

  const int lane  = threadIdx.x & 31;
  const int wave  = threadIdx.x >> 5;
  const int m16   = lane & 15;
  const int khalf = lane >> 4;

  // Issue async copies of one 64x64 K tile + V tile into LDS buffer `buf`.
  // 1024 16B-chunks per array / 128 threads = 8 B128 issues each, per array.
  auto stage = [&](int buf, int j0) {
#pragma unroll
    for (int q = 0; q < 8; ++q) {
      const int idx = q * 128 + (int)threadIdx.x;   // 0..1023
      const int c  = idx >> 4;
      const int j4 = (idx & 15) * 4;
      async_copy_b128(&Kb[(size_t)c * kN + j0 + j4], &ks[buf][c][j4]);
      async_copy_b128(&Vb[(size_t)c * kN + j0 + j4], &vs[buf][c][j4]);
    }
  };

  // Resident A fragments of Q^T for this wave's 16 query rows (32 VGPRs).
  v2f aQ[16];
  {
    const int irow = i0 + wave * 16 + m16;
#pragma unroll
    for (int kk = 0; kk < 16; ++kk) {
      const int kidx = kk * 4 + khalf * 2;
      aQ[kk].x = Qb[(size_t)kidx * kN + irow];
      aQ[kk].y = Qb[(size_t)(kidx + 1) * kN + irow];
    }
  }

  v8f   acc[4] = {v8f{}, v8f{}, v8f{}, v8f{}};  // [i(16)][c(64)] per wave
  float rsum[8] = {};                           // softmax denominators

  stage(0, 0);                                  // prologue prefetch

  constexpr int kTiles = kN / kTile;            // 100
  for (int t = 0; t < kTiles; ++t) {
    const int buf = t & 1;
    wait_async_zero();     // tile t landed in ks/vs[buf]
    __syncthreads();       // ...for every wave; buf^1 readers also done
    if (t + 1 < kTiles) stage(buf ^ 1, (t + 1) * kTile);  // overlap next tile

    const int j0 = t * kTile;

    // ---- S = Q^T K tile, then P = exp(S * mask_j) ------------------------
#pragma unroll
    for (int jt = 0; jt < 4; ++jt) {
      v8f s = {};
#pragma unroll
      for (int kk = 0; kk < 16; ++kk) {
        const int kidx = kk * 4 + khalf * 2;
        const int jl   = jt * 16 + m16;
        v2f bf;
        bf.x = ks[buf][kidx][jl];
        bf.y = ks[buf][kidx + 1][jl];
        s = wmma_f32(aQ[kk], bf, s);
      }
      // mask value depends only on column -> one load per lane per jt
      const float mval = mb[j0 + jt * 16 + m16];
#pragma unroll
      for (int r = 0; r < 8; ++r) {
        const float p = __expf(s[r] * mval);
        ss[wave * 16 + r + khalf * 8][jt * 16 + m16] = p;
        // row-sum: reduce over the 16 lanes sharing this row (xor 1,2,4,8
        // stays inside each 16-lane half of the wave32)
        float tt = p;
        tt += __shfl_xor(tt, 1);
        tt += __shfl_xor(tt, 2);
        tt += __shfl_xor(tt, 4);
        tt += __shfl_xor(tt, 8);
        rsum[r] += tt;
      }
    }

    // ---- acc += P * V^T  (K-dim = j, 16 steps of 4) ----------------------
    // Each wave reads only the ss rows it wrote: same-wave LDS ops are
    // in-order, no workgroup barrier needed here.
#pragma unroll
    for (int ct = 0; ct < 4; ++ct) {
#pragma unroll
      for (int kk = 0; kk < 16; ++kk) {
        const int jl = kk * 4 + khalf * 2;
        v2f aP, bV;
        aP.x = ss[wave * 16 + m16][jl];
        aP.y = ss[wave * 16 + m16][jl + 1];
        bV.x = vs[buf][ct * 16 + m16][jl];       // B[k=j][n=c] = V[c][j]
        bV.y = vs[buf][ct * 16 + m16][jl + 1];
        acc[ct] = wmma_f32(aP, bV, acc[ct]);
      }
    }
  }

  // ---- finalize: agg = acc / rsum, blend, coalesced store via LDS --------
  __syncthreads();
#pragma unroll
  for (int ct = 0; ct < 4; ++ct) {
#pragma unroll
    for (int r = 0; r < 8; ++r) {
      const int il = wave * 16 + r + khalf * 8;
      ss[il][ct * 16 + m16] = acc[ct][r] / rsum[r];   // ss[i][c] = agg
    }
  }
  __syncthreads();

  const float g = gamma[0];
  const float* fgb  = fg  + (size_t)b * kC * kN;
  float*       outb = (float*)out + (size_t)b * kC * kN;
  for (int idx = threadIdx.x; idx < kTile * kTile; idx += 128) {
    const int c = idx >> 6, i = idx & 63;        // consecutive i -> coalesced
    const float mv = mb[i0 + i];
    const float f  = fgb[(size_t)c * kN + i0 + i];
    outb[(size_t)c * kN + i0 + i] = f * mv + g * (1.0f - mv) * ss[i][c];
  }
}

// ---------------------------------------------------------------------------
extern "C" void kernel_launch(void* const* d_in, const int* in_sizes, int n_in,
                              void* d_out, int out_size, void* d_ws,
                              size_t ws_size, hipStream_t stream) {
  (void)in_sizes; (void)n_in; (void)out_size; (void)ws_size;
  const float* fg    = (const float*)d_in[0];
  const float* bg    = (const float*)d_in[1];
  const float* mask  = (const float*)d_in[2];
  const float* wq    = (const float*)d_in[3];
  const float* bq    = (const float*)d_in[4];
  const float* wk    = (const float*)d_in[5];
  const float* bk    = (const float*)d_in[6];
  const float* wv    = (const float*)d_in[7];
  const float* bv    = (const float*)d_in[8];
  const float* gamma = (const float*)d_in[9];
  float* out = (float*)d_out;

  // Workspace: Q,K,V each [2][64][6400] f32 -> 3 * 3.28 MB = 9.83 MB.
  float* Q = (float*)d_ws;
  float* K = Q + (size_t)2 * kC * kN;
  float* V = K + (size_t)2 * kC * kN;

  dim3 g1(kN / kTile, 2, 3);
  qkv_proj_kernel<<<g1, 128, 0, stream>>>(fg, bg, wq, bq, wk, bk, wv, bv,
                                          Q, K, V);
  dim3 g2(kN / kTile, 2);
  flash_attn_kernel<<<g2, 128, 0, stream>>>(Q, K, V, mask, fg, gamma, out);
}